// BetterBetterTorchLayer_26362509263439
// MI455X (gfx1250) — compile-verified
//
#include <hip/hip_runtime.h>
#include <math.h>

// ---------------------------------------------------------------------------
// 16-qubit statevector sim on gfx1250 (MI455X), wave32 + V_WMMA_F32_16X16X4_F32
//
//  - encoder (48 RY + 120 CRZ) folded analytically into one init pass
//  - each layer's RX/RY/RZ fused into 4 complex 16x16 matrices (4 qubits each),
//    applied in-place as WMMA tensor contractions (16 wmma ops / tile)
//  - each layer's 16-CNOT ring composed on host into a GF(2) linear map,
//    applied as a single gather pass (popcount parity per output bit)
// ---------------------------------------------------------------------------

typedef float v2f __attribute__((ext_vector_type(2)));
typedef float v8f __attribute__((ext_vector_type(8)));

#define NQ    16
#define DIM   65536          // 2^16 amplitudes
#define NB    8              // batch
#define NEDGE 121

struct Masks { unsigned m[16]; };   // output bit p = parity(j & m[p])

// ---- build the 8 fused 16x16 complex gate matrices (2 layers x 4 groups) ----
// gates[((l*4+g)*16 + k')*16 + k] = tensor product of u_q over the 4 bits of
// the group; group g covers absolute bit positions 4g..4g+3 (qubit = 15 - pos).
__global__ __launch_bounds__(256) void build_gates(const float* __restrict__ lp,
                                                   float2* __restrict__ gates) {
  int tid = threadIdx.x;
  for (int e = 0; e < 8; ++e) {
    int idx = e * 256 + tid;          // 0..2047
    int lg  = idx >> 8;               // 0..7
    int l = lg >> 2, g = lg & 3;
    int ent = idx & 255;
    int kp = ent >> 4, k = ent & 15;
    float re = 1.f, im = 0.f;
    for (int m = 0; m < 4; ++m) {
      int p = 4 * g + m;              // absolute bit position
      int q = 15 - p;                 // qubit index (wire 0 = MSB)
      float a = lp[(l * 3 + 0) * NQ + q];   // RX angle
      float b = lp[(l * 3 + 1) * NQ + q];   // RY angle
      float c = lp[(l * 3 + 2) * NQ + q];   // RZ angle
      float cx = cosf(0.5f * a), sx = sinf(0.5f * a);
      float cy = cosf(0.5f * b), sy = sinf(0.5f * b);
      float cz = cosf(0.5f * c), sz = sinf(0.5f * c);
      // M1 = RY * RX
      float m1r[2][2], m1i[2][2];
      m1r[0][0] =  cy * cx;  m1i[0][0] =  sy * sx;
      m1r[0][1] = -sy * cx;  m1i[0][1] = -cy * sx;
      m1r[1][0] =  sy * cx;  m1i[1][0] = -cy * sx;
      m1r[1][1] =  cy * cx;  m1i[1][1] = -sy * sx;
      int br = (kp >> m) & 1, bc = (k >> m) & 1;
      float r0 = m1r[br][bc], i0 = m1i[br][bc];
      // u = RZ * M1 : row0 *= e^{-i c/2}, row1 *= e^{+i c/2}
      float ur, ui;
      if (br == 0) { ur = r0 * cz + i0 * sz; ui = i0 * cz - r0 * sz; }
      else         { ur = r0 * cz - i0 * sz; ui = i0 * cz + r0 * sz; }
      float nr = re * ur - im * ui;
      float ni = re * ui + im * ur;
      re = nr; im = ni;
    }
    float2 o; o.x = re; o.y = im;
    gates[idx] = o;
  }
}

// ---- analytic post-encoder state: product-state magnitudes x CRZ phases ----
__global__ __launch_bounds__(256) void init_state(const float* __restrict__ node,
                                                  const float* __restrict__ edge,
                                                  float2* __restrict__ S) {
  __shared__ float cw[NQ], sw[NQ], t[NEDGE];
  int b   = blockIdx.x >> 8;
  int tid = threadIdx.x;
  if (tid < NQ) {
    float th = node[(b * NQ + tid) * 3 + 0] +
               node[(b * NQ + tid) * 3 + 1] +
               node[(b * NQ + tid) * 3 + 2];        // RY angles add
    cw[tid] = cosf(0.5f * th);
    sw[tid] = sinf(0.5f * th);
  }
  for (int j = tid; j < NEDGE; j += 256) t[j] = 0.5f * edge[b * NEDGE + j];
  __syncthreads();
  int i = (blockIdx.x & 255) * 256 + tid;
  float mag = 1.f;
  #pragma unroll
  for (int q = 0; q < NQ; ++q)
    mag *= ((i >> (15 - q)) & 1) ? sw[q] : cw[q];
  float ph = 0.f;
  int idx = 1;                                      // col 0 unused (source bug)
  for (int q = 0; q < NQ; ++q) {
    int bq = (i >> (15 - q)) & 1;
    for (int e = q + 1; e < NQ; ++e) {
      if (bq) ph += ((i >> (15 - e)) & 1) ? t[idx] : -t[idx];
      ++idx;
    }
  }
  float2 a; a.x = mag * cosf(ph); a.y = mag * sinf(ph);
  S[b * DIM + i] = a;
}

// ---- apply one 16x16 complex gate along bit-field [p..p+3], in place --------
// One full wave per 16-column tile. Complex matmul = 4 real f32 WMMAs, each
// accumulated over K=16 in 4 chunks of K=4 (V_WMMA_F32_16X16X4_F32).
__global__ __launch_bounds__(256) void apply_group(float2* __restrict__ S,
                                                   const float2* __restrict__ gates,
                                                   int p) {
  int lane = threadIdx.x & 31;
  int wv   = blockIdx.x * (blockDim.x >> 5) + (threadIdx.x >> 5);
  int b    = wv >> 8;           // batch
  int T    = wv & 255;          // column tile
  int n    = lane & 15;
  int hh   = lane >> 4;         // lane half
  unsigned c  = (unsigned)T * 16 + n;
  unsigned lo = c & ((1u << p) - 1u);
  unsigned hi = c >> p;
  unsigned colbase = (hi << (p + 4)) | lo;   // add (k << p) per element
  float2* Sb = S + (size_t)b * DIM;

  // A fragment (ISA layout): lane m=lane&15; VGPR0 K = 4kk + (hh?2:0), VGPR1 +1
  int m = lane & 15;
  v2f Ar[4], Ai[4], AiN[4];
  #pragma unroll
  for (int kk = 0; kk < 4; ++kk) {
    int k0 = 4 * kk + (hh ? 2 : 0);
    float2 g0 = gates[m * 16 + k0];
    float2 g1 = gates[m * 16 + k0 + 1];
    Ar[kk][0]  = g0.x;  Ar[kk][1]  = g1.x;
    Ai[kk][0]  = g0.y;  Ai[kk][1]  = g1.y;
    AiN[kk][0] = -g0.y; AiN[kk][1] = -g1.y;   // f32 WMMA has no A-neg modifier
  }
  // B fragment (rows striped across lanes): VGPR0 k = 4kk+(hh?1:0), VGPR1 +2
  v2f Br[4], Bi[4];
  #pragma unroll
  for (int kk = 0; kk < 4; ++kk) {
    int kb0 = 4 * kk + (hh ? 1 : 0);
    int kb1 = kb0 + 2;
    float2 s0 = Sb[colbase | ((unsigned)kb0 << p)];
    float2 s1 = Sb[colbase | ((unsigned)kb1 << p)];
    Br[kk][0] = s0.x; Br[kk][1] = s1.x;
    Bi[kk][0] = s0.y; Bi[kk][1] = s1.y;
  }

  v8f Dr = {}; v8f Di = {};
  #pragma unroll
  for (int kk = 0; kk < 4; ++kk)   // Re: +Ur*Sr
    Dr = __builtin_amdgcn_wmma_f32_16x16x4_f32(false, Ar[kk], false, Br[kk],
                                               (short)0, Dr, false, false);
  #pragma unroll
  for (int kk = 0; kk < 4; ++kk)   // Re: -Ui*Si
    Dr = __builtin_amdgcn_wmma_f32_16x16x4_f32(false, AiN[kk], false, Bi[kk],
                                               (short)0, Dr, false, false);
  #pragma unroll
  for (int kk = 0; kk < 4; ++kk)   // Im: +Ur*Si
    Di = __builtin_amdgcn_wmma_f32_16x16x4_f32(false, Ar[kk], false, Bi[kk],
                                               (short)0, Di, false, false);
  #pragma unroll
  for (int kk = 0; kk < 4; ++kk)   // Im: +Ui*Sr
    Di = __builtin_amdgcn_wmma_f32_16x16x4_f32(false, Ai[kk], false, Br[kk],
                                               (short)0, Di, false, false);

  // D layout: VGPR v -> row m' = v + (hh?8:0), col = lane&15
  #pragma unroll
  for (int v = 0; v < 8; ++v) {
    unsigned mp = (unsigned)v + (hh ? 8u : 0u);
    float2 o; o.x = Dr[v]; o.y = Di[v];
    Sb[colbase | (mp << p)] = o;
  }
}

// ---- CNOT ring as a GF(2) linear gather: out[j] = in[src(j)] ---------------
__global__ __launch_bounds__(256) void cnot_perm(const float2* __restrict__ in,
                                                 float2* __restrict__ out,
                                                 Masks mk) {
  int b = blockIdx.x >> 8;
  int j = (blockIdx.x & 255) * 256 + threadIdx.x;
  unsigned src = 0;
  #pragma unroll
  for (int pbit = 0; pbit < 16; ++pbit)
    src |= (unsigned)(__popc((unsigned)j & mk.m[pbit]) & 1) << pbit;
  out[(size_t)b * DIM + j] = in[(size_t)b * DIM + src];
}

// ---- |amp|^2 ----------------------------------------------------------------
__global__ __launch_bounds__(256) void probs(const float2* __restrict__ S,
                                             float* __restrict__ out) {
  int idx = blockIdx.x * 256 + threadIdx.x;
  float2 a = S[idx];
  out[idx] = a.x * a.x + a.y * a.y;
}

extern "C" void kernel_launch(void* const* d_in, const int* in_sizes, int n_in,
                              void* d_out, int out_size, void* d_ws, size_t ws_size,
                              hipStream_t stream) {
  (void)in_sizes; (void)n_in; (void)out_size; (void)ws_size;
  const float* node = (const float*)d_in[0];
  // d_in[1] = node_indices (int64) — unused by the reference
  const float* edge = (const float*)d_in[2];
  const float* lp   = (const float*)d_in[3];

  float2* S0    = (float2*)d_ws;          // NB*DIM float2 = 4 MB
  float2* S1    = S0 + (size_t)NB * DIM;  // 4 MB (CNOT gather target)
  float2* gates = S1 + (size_t)NB * DIM;  // 2*4*256 float2 = 16 KB

  build_gates<<<1, 256, 0, stream>>>(lp, gates);
  init_state<<<NB * 256, 256, 0, stream>>>(node, edge, S0);

  float2* cur = S0;
  float2* oth = S1;
  for (int l = 0; l < 2; ++l) {
    for (int g = 0; g < 4; ++g)
      apply_group<<<256, 256, 0, stream>>>(cur, gates + (size_t)(l * 4 + g) * 256, 4 * g);

    // Compose the 16-CNOT ring into a single linear gather map.
    Masks mk;
    for (int pp = 0; pp < 16; ++pp) mk.m[pp] = 1u << pp;
    int jump = (l + 1) % (NQ - 1);
    for (int k = NQ - 1; k >= 0; --k) {       // reverse order: src = pi1∘…∘pi16
      int cq = k, tq = (k + jump) % NQ;
      int pc = 15 - cq, pt = 15 - tq;         // wire 0 = most significant bit
      mk.m[pt] ^= mk.m[pc];
    }
    cnot_perm<<<NB * 256, 256, 0, stream>>>(cur, oth, mk);
    float2* tmp = cur; cur = oth; oth = tmp;
  }

  probs<<<NB * 256, 256, 0, stream>>>(cur, (float*)d_out);
}